// LabeledChamferDistance_86517821214291
// MI455X (gfx1250) — compile-verified
//
#include <hip/hip_runtime.h>

typedef __attribute__((ext_vector_type(2))) float v2f;
typedef __attribute__((ext_vector_type(8))) float v8f;

#define BATCH 8
#define NPTS  4096
#define TILES (NPTS / 16)   // 256 column tiles of 16

#define BETA  1.0f
#define GAMMA 1.0f
#define DELTA 0.0f

// One wave handles a 16-row tile of X against all of Y (one batch).
// WMMA computes e[m][n] = |y_n|^2 - 2*x_m.y_n directly by folding constants
// into the K=4 slot:  A[m][*] = (-2x0,-2x1,-2x2, 1),  B[*][n] = (y0,y1,y2,|y|^2).
// Then d[m][n] = e[m][n] + |x_m|^2, and |x_m|^2 is row-constant, so min/argmin
// can run on e alone; |x_m|^2 is added back before the cross-lane reduction.
__global__ __launch_bounds__(256) void chamfer_nn_wmma(
    const float* __restrict__ X, const float* __restrict__ Y,
    float* __restrict__ minDist, float* __restrict__ minIdxF)
{
    const int lane        = threadIdx.x & 31;
    const int waveInBlock = threadIdx.x >> 5;
    const int waveGlobal  = blockIdx.x * (blockDim.x >> 5) + waveInBlock;
    const int b           = waveGlobal / TILES;
    const int tileRow     = waveGlobal % TILES;
    const int rowBase     = tileRow * 16;

    const float* Xb = X + (size_t)b * NPTS * 3;
    const float* Yb = Y + (size_t)b * NPTS * 3;

    const int  half = lane >> 4;    // 0: lanes 0-15 (K=0,1 / rows 0-7), 1: lanes 16-31 (K=2,3 / rows 8-15)
    const int  l16  = lane & 15;
    const bool hi   = (half != 0);

    // ---- A matrix (branchless): every lane loads its full row, then selects.
    // ISA layout: v0: lanes0-15 K=0, lanes16-31 K=2 ; v1: lanes0-15 K=1, lanes16-31 K=3
    v2f  A;
    float rowsq;
    {
        const float* p = Xb + (size_t)(rowBase + l16) * 3;
        float a0 = p[0], a1 = p[1], a2 = p[2];            // contiguous 12B -> b96
        rowsq = a0 * a0 + a1 * a1 + a2 * a2;              // |x_row|^2 (row = l16)
        A.x = hi ? (-2.0f * a2) : (-2.0f * a0);           // K=2 : K=0
        A.y = hi ? 1.0f         : (-2.0f * a1);           // K=3 : K=1
    }

    // per-row |x|^2 for the 8 rows this half-wave's D-registers own (8 bpermutes, once)
    float xsq[8];
    #pragma unroll
    for (int j = 0; j < 8; ++j)
        xsq[j] = __shfl(rowsq, half * 8 + j, 32);         // rows 0-15 live in lanes 0-15

    float bestE[8];
    int   bestIdx[8];
    #pragma unroll
    for (int j = 0; j < 8; ++j) { bestE[j] = 3.402823466e38f; bestIdx[j] = 0; }

    // one 16-column tile: ysq, B-select, WMMA, min/argmin update
    auto process = [&](float c0, float c1, float c2, int col) {
        const float ysq = c0 * c0 + c1 * c1 + c2 * c2;
        v2f Bm;
        Bm.x = hi ? c2  : c0;                             // K=2 : K=0
        Bm.y = hi ? ysq : c1;                             // K=3 : K=1

        v8f acc = {};
        // acc[j] = |y_col|^2 - 2 * x_(rowBase + half*8 + j) . y_col
        acc = __builtin_amdgcn_wmma_f32_16x16x4_f32(
                  /*neg_a=*/false, A, /*neg_b=*/false, Bm,
                  /*c_mod=*/(short)0, acc, /*reuse_a=*/false, /*reuse_b=*/false);

        #pragma unroll
        for (int j = 0; j < 8; ++j) {
            float e = acc[j];
            if (e < bestE[j]) { bestE[j] = e; bestIdx[j] = col; }   // strict <: first hit wins
        }
    };

    // ---- software-pipelined column sweep with peeled tail:
    // steady-state loop prefetches unconditionally via pointer bump (pY += 48 floats),
    // so no per-iteration address recomputation; last tile consumed after the loop.
    const float* pY = Yb + (size_t)l16 * 3;               // tile 0, column l16
    float n0 = pY[0], n1 = pY[1], n2 = pY[2];
    int col = l16;

    for (int t = 0; t < TILES - 1; ++t) {
        const float c0 = n0, c1 = n1, c2 = n2;
        pY += 48;                                          // next tile (16 cols * 3 floats)
        n0 = pY[0]; n1 = pY[1]; n2 = pY[2];               // contiguous 12B -> b96 (in-bounds: <= tile 255)
        process(c0, c1, c2, col);
        col += 16;
    }
    process(n0, n1, n2, col);                             // tile 255

    // ---- branchless cross-lane (min, argmin) over the 16 lanes of each half-group.
    // d >= 0 so f32 bits order as u32; pack (d_bits << 32) | idx and u64-min.
    // Equal d -> smaller idx wins (jnp.argmin first-occurrence semantics).
    #pragma unroll
    for (int j = 0; j < 8; ++j) {
        const float dj = bestE[j] + xsq[j];               // true distance for this row
        unsigned long long key =
            ((unsigned long long)__float_as_uint(dj) << 32) | (unsigned)bestIdx[j];
        #pragma unroll
        for (int off = 1; off < 16; off <<= 1) {
            unsigned long long ok = __shfl_xor(key, off, 32);
            key = (ok < key) ? ok : key;
        }
        if (l16 == 0) {   // lane 0 writes rows 0-7, lane 16 writes rows 8-15
            const int    row = rowBase + half * 8 + j;
            const size_t o   = (size_t)b * NPTS + row;
            minDist[o] = __uint_as_float((unsigned)(key >> 32));
            minIdxF[o] = (float)(unsigned)(key & 0xFFFFFFFFu);
        }
    }
}

// Deterministic final reduction: one wave per batch, then thread 0 combines.
__global__ __launch_bounds__(256) void chamfer_loss(
    const float* __restrict__ dist12, const float* __restrict__ dist21,
    float* __restrict__ out)
{
    __shared__ float sSum12[BATCH], sMax12[BATCH], sSum21[BATCH];
    const int wave = threadIdx.x >> 5;   // == batch (8 waves)
    const int lane = threadIdx.x & 31;

    const float* p12 = dist12 + (size_t)wave * NPTS;
    const float* p21 = dist21 + (size_t)wave * NPTS;

    float s12 = 0.0f, s21 = 0.0f, m12 = -3.402823466e38f;
    for (int i = lane; i < NPTS; i += 32) {
        float v = p12[i];
        s12 += v;
        m12  = fmaxf(m12, v);
        s21 += p21[i];
    }
    #pragma unroll
    for (int off = 16; off > 0; off >>= 1) {
        s12 += __shfl_xor(s12, off, 32);
        s21 += __shfl_xor(s21, off, 32);
        m12  = fmaxf(m12, __shfl_xor(m12, off, 32));
    }
    if (lane == 0) { sSum12[wave] = s12; sMax12[wave] = m12; sSum21[wave] = s21; }
    __syncthreads();

    if (threadIdx.x == 0) {
        float loss = 0.0f;
        for (int b = 0; b < BATCH; ++b) {
            loss += sSum12[b] / (float)NPTS
                  + BETA * sMax12[b]
                  + (GAMMA + DELTA * (float)NPTS) * (sSum21[b] / (float)NPTS);
        }
        out[0] = loss / (float)BATCH;
    }
}

extern "C" void kernel_launch(void* const* d_in, const int* in_sizes, int n_in,
                              void* d_out, int out_size, void* d_ws, size_t ws_size,
                              hipStream_t stream) {
    (void)in_sizes; (void)n_in; (void)out_size; (void)ws_size;

    const float* xyz1 = (const float*)d_in[0];
    const float* xyz2 = (const float*)d_in[1];
    float* out = (float*)d_out;
    float* ws  = (float*)d_ws;

    float* dist12 = ws;                         // [8,4096]
    float* dist21 = ws + (size_t)BATCH * NPTS;  // [8,4096]
    float* idx12  = out + 1;                    // [8,4096] as float
    float* idx21  = out + 1 + (size_t)BATCH * NPTS;

    dim3 blk(256);                              // 8 waves/block
    dim3 grd((BATCH * TILES) / 8);              // 2048 waves -> 256 blocks

    chamfer_nn_wmma<<<grd, blk, 0, stream>>>(xyz1, xyz2, dist12, idx12);
    chamfer_nn_wmma<<<grd, blk, 0, stream>>>(xyz2, xyz1, dist21, idx21);
    chamfer_loss<<<1, 256, 0, stream>>>(dist12, dist21, out);
}